// GATModel_54271206752836
// MI455X (gfx1250) — compile-verified
//
#include <hip/hip_runtime.h>
#include <hip/hip_bf16.h>
#include <math.h>

// ---------------------------------------------------------------------------
// GAT (3 layers, heads 4/4/1) for MI455X (gfx1250, wave32).
// Dense GEMMs use V_WMMA_F32_16X16X4_F32 (fp32 matrix cores, exact fp32 path).
// Edge softmax/aggregation is the memory-bound part: VMEM + f32 atomics.
// ---------------------------------------------------------------------------

#define NN 100000
#define EE 1000000
#define DD 64
#define ETOT (EE + NN)

typedef float v2f __attribute__((ext_vector_type(2)));
typedef float v8f __attribute__((ext_vector_type(8)));

// ---- order-preserving float <-> uint encoding for atomic max ----
__device__ __forceinline__ unsigned enc_f(float f) {
    unsigned u = __float_as_uint(f);
    return (u & 0x80000000u) ? ~u : (u | 0x80000000u);
}
__device__ __forceinline__ float dec_f(unsigned u) {
    u = (u & 0x80000000u) ? (u & 0x7FFFFFFFu) : ~u;
    return __uint_as_float(u);
}

// ---- fills ----
__global__ void fill_f32(float* p, float v, size_t n) {
    size_t i = (size_t)blockIdx.x * blockDim.x + threadIdx.x;
    if (i < n) p[i] = v;
}
__global__ void fill_u32(unsigned* p, unsigned v, size_t n) {
    size_t i = (size_t)blockIdx.x * blockDim.x + threadIdx.x;
    if (i < n) p[i] = v;
}

// ---- self-loop attr: deg + segment_sum(edge_attr) by dst ----
__global__ void deg_sum_k(const int* __restrict__ ei, const float* __restrict__ ea,
                          float* __restrict__ deg, float* __restrict__ lsum) {
    int e = blockIdx.x * blockDim.x + threadIdx.x;
    if (e >= EE) return;
    int t = ei[EE + e];
    atomicAdd(deg + t, 1.0f);
    atomicAdd(lsum + (size_t)t * 2 + 0, ea[(size_t)e * 2 + 0]);
    atomicAdd(lsum + (size_t)t * 2 + 1, ea[(size_t)e * 2 + 1]);
}
__global__ void loop_fin_k(float* __restrict__ loop_attr, const float* __restrict__ deg) {
    int n = blockIdx.x * blockDim.x + threadIdx.x;
    if (n >= NN) return;
    float dv = deg[n]; dv = dv > 1.0f ? dv : 1.0f;
    loop_attr[(size_t)n * 2 + 0] /= dv;
    loop_attr[(size_t)n * 2 + 1] /= dv;
}

// ---- encoder: h = relu(x @ enc_W + enc_b), x:[N,2], enc_W:[2,64] ----
__global__ void encoder_k(const float* __restrict__ x, const float* __restrict__ W,
                          const float* __restrict__ b, float* __restrict__ h) {
    int idx = blockIdx.x * blockDim.x + threadIdx.x;
    if (idx >= NN * DD) return;
    int n = idx >> 6, d = idx & 63;
    float v = x[(size_t)n * 2] * W[d] + x[(size_t)n * 2 + 1] * W[DD + d] + b[d];
    h[idx] = v > 0.0f ? v : 0.0f;
}

// ---- WMMA f32 GEMM: C[N, M] = A[N,64] @ B[64,M]  (one 16x16 tile per wave) ----
// Optional epilogue: +bias, relu (for fc1).
__global__ void gemm_wmma_f32(const float* __restrict__ A, const float* __restrict__ B,
                              float* __restrict__ C, const float* __restrict__ bias,
                              int M, int numColTiles, int totalTiles, int doRelu) {
    int wave = (int)((blockIdx.x * blockDim.x + threadIdx.x) >> 5);  // wave32
    int lane = threadIdx.x & 31;
    if (wave >= totalTiles) return;  // wave-uniform: EXEC stays all-1s inside active waves
    int rowTile = wave / numColTiles;
    int colTile = wave % numColTiles;
    int r0 = rowTile * 16;
    int c0 = colTile * 16;
    int lr = lane & 15;   // A: row-in-tile; B/C: col-in-tile
    int hi = lane >> 4;   // 0 -> K pair {0,1}, 1 -> K pair {2,3}

    v8f acc = {};
    const float* Arow = A + (size_t)(r0 + lr) * DD;
    const float* Bcol = B + (size_t)(c0 + lr);
#pragma unroll
    for (int k0 = 0; k0 < DD; k0 += 4) {
        int kb = k0 + hi * 2;
        v2f a, b;
        a.x = Arow[kb];
        a.y = Arow[kb + 1];
        b.x = Bcol[(size_t)kb * M];
        b.y = Bcol[(size_t)(kb + 1) * M];
        // D = A(16x4) * B(4x16) + C ; 8 args: neg_a, A, neg_b, B, c_mod, C, reuse_a, reuse_b
        acc = __builtin_amdgcn_wmma_f32_16x16x4_f32(false, a, false, b, (short)0, acc,
                                                    false, false);
    }
#pragma unroll
    for (int i = 0; i < 8; ++i) {
        int m = hi * 8 + i;
        float v = acc[i];
        if (doRelu) { v += bias[c0 + lr]; v = v > 0.0f ? v : 0.0f; }
        C[(size_t)(r0 + m) * M + c0 + lr] = v;
    }
}

// ---- a_src/a_dst from xw: a[n,h] = sum_d xw[n,h,d]*att[h,d] ----
__global__ void att_node_k(const float* __restrict__ xw, const float* __restrict__ att_s,
                           const float* __restrict__ att_d, float* __restrict__ a_src,
                           float* __restrict__ a_dst, int H) {
    int idx = blockIdx.x * blockDim.x + threadIdx.x;
    if (idx >= NN * H) return;
    int n = idx / H, h = idx % H;
    const float* row = xw + (size_t)n * H * DD + h * DD;
    float ss = 0.0f, sd = 0.0f;
#pragma unroll 8
    for (int d = 0; d < DD; ++d) {
        float v = row[d];
        ss += v * att_s[h * DD + d];
        sd += v * att_d[h * DD + d];
    }
    a_src[idx] = ss;
    a_dst[idx] = sd;
}

// ---- fold a_e: coef[k,h] = sum_d edge_W[k, h*64+d] * att_edge[h,d] ----
__global__ void edge_coef_k(const float* __restrict__ edge_W, const float* __restrict__ att_e,
                            float* __restrict__ coef, int H) {
    int idx = threadIdx.x;
    if (idx >= 2 * H) return;
    int k = idx / H, h = idx % H;
    float s = 0.0f;
    for (int d = 0; d < DD; ++d)
        s += edge_W[(size_t)k * H * DD + h * DD + d] * att_e[h * DD + d];
    coef[k * H + h] = s;
}

// ---- pass 1: logits + segment max ----
__global__ void edge_logits_k(const int* __restrict__ ei, const float* __restrict__ ea,
                              const float* __restrict__ loop_attr,
                              const float* __restrict__ a_src, const float* __restrict__ a_dst,
                              const float* __restrict__ coef, float* __restrict__ logits,
                              unsigned* __restrict__ menc, int H) {
    int e = blockIdx.x * blockDim.x + threadIdx.x;
    if (e >= ETOT) return;
    int s, t; float e0, e1;
    if (e < EE) {
        s = ei[e]; t = ei[EE + e];
        e0 = ea[(size_t)e * 2]; e1 = ea[(size_t)e * 2 + 1];
    } else {
        s = t = e - EE;
        e0 = loop_attr[(size_t)s * 2]; e1 = loop_attr[(size_t)s * 2 + 1];
    }
    for (int h = 0; h < H; ++h) {
        float lg = a_src[(size_t)s * H + h] + a_dst[(size_t)t * H + h]
                 + e0 * coef[h] + e1 * coef[H + h];
        lg = lg > 0.0f ? lg : 0.2f * lg;  // leaky_relu(0.2)
        logits[(size_t)e * H + h] = lg;
        atomicMax(menc + (size_t)t * H + h, enc_f(lg));
    }
}

// ---- pass 2: exp(logit - max) + segment sum ----
__global__ void edge_exp_k(const int* __restrict__ ei, float* __restrict__ logits,
                           const unsigned* __restrict__ menc, float* __restrict__ den, int H) {
    int e = blockIdx.x * blockDim.x + threadIdx.x;
    if (e >= ETOT) return;
    int t = (e < EE) ? ei[EE + e] : (e - EE);
    for (int h = 0; h < H; ++h) {
        float lg = logits[(size_t)e * H + h];
        float mm = dec_f(menc[(size_t)t * H + h]);
        float ex = __expf(lg - mm);
        logits[(size_t)e * H + h] = ex;
        atomicAdd(den + (size_t)t * H + h, ex);
    }
}

// ---- pass 3: out[dst,h,:] += alpha * xw[src,h,:] ----
__global__ void edge_aggr_k(const int* __restrict__ ei, const float* __restrict__ ex,
                            const float* __restrict__ den, const float* __restrict__ xw,
                            float* __restrict__ outbuf, int H) {
    long long idx = (long long)blockIdx.x * blockDim.x + threadIdx.x;
    if (idx >= (long long)ETOT * H) return;
    int e = (int)(idx / H), h = (int)(idx % H);
    int s, t;
    if (e < EE) { s = ei[e]; t = ei[EE + e]; } else { s = t = e - EE; }
    float alpha = ex[(size_t)e * H + h] / (den[(size_t)t * H + h] + 1e-16f);
    const float* xs = xw + (size_t)s * H * DD + h * DD;
    float* od = outbuf + (size_t)t * H * DD + h * DD;
#pragma unroll
    for (int d = 0; d < DD; d += 4) {
        float4 v = *(const float4*)(xs + d);
        atomicAdd(od + d + 0, v.x * alpha);
        atomicAdd(od + d + 1, v.y * alpha);
        atomicAdd(od + d + 2, v.z * alpha);
        atomicAdd(od + d + 3, v.w * alpha);
    }
}

// ---- h_new = relu(mean_h(out) + bias) + h_old ----
__global__ void layer_fin_k(const float* __restrict__ outbuf, const float* __restrict__ bias,
                            const float* __restrict__ hin, float* __restrict__ hout, int H) {
    int idx = blockIdx.x * blockDim.x + threadIdx.x;
    if (idx >= NN * DD) return;
    int n = idx >> 6, d = idx & 63;
    float acc = 0.0f;
    for (int h = 0; h < H; ++h) acc += outbuf[(size_t)n * H * DD + h * DD + d];
    acc = acc / (float)H + bias[d];
    acc = acc > 0.0f ? acc : 0.0f;
    hout[idx] = acc + hin[idx];
}

// ---- fc2 + tanh ----
__global__ void fc2_k(const float* __restrict__ h2, const float* __restrict__ W,
                      const float* __restrict__ b, float* __restrict__ out) {
    int n = blockIdx.x * blockDim.x + threadIdx.x;
    if (n >= NN) return;
    float o0 = b[0], o1 = b[1];
#pragma unroll
    for (int j = 0; j < 32; ++j) {
        float v = h2[(size_t)n * 32 + j];
        o0 += v * W[j * 2 + 0];
        o1 += v * W[j * 2 + 1];
    }
    out[(size_t)n * 2 + 0] = tanhf(o0);
    out[(size_t)n * 2 + 1] = tanhf(o1);
}

static inline int gridFor(long long n, int bs) { return (int)((n + bs - 1) / bs); }

extern "C" void kernel_launch(void* const* d_in, const int* in_sizes, int n_in,
                              void* d_out, int out_size, void* d_ws, size_t ws_size,
                              hipStream_t stream) {
    // ---- inputs (setup_inputs order) ----
    const float* x         = (const float*)d_in[0];
    const int*   ei        = (const int*)  d_in[1];   // [2,E]
    const float* ea        = (const float*)d_in[2];   // [E,2]
    const float* enc_W     = (const float*)d_in[3];
    const float* enc_b     = (const float*)d_in[4];
    const float* lin_W[3]  = { (const float*)d_in[5],  (const float*)d_in[11], (const float*)d_in[17] };
    const float* att_src[3]= { (const float*)d_in[6],  (const float*)d_in[12], (const float*)d_in[18] };
    const float* att_dst[3]= { (const float*)d_in[7],  (const float*)d_in[13], (const float*)d_in[19] };
    const float* edge_W[3] = { (const float*)d_in[8],  (const float*)d_in[14], (const float*)d_in[20] };
    const float* att_edge[3]={ (const float*)d_in[9],  (const float*)d_in[15], (const float*)d_in[21] };
    const float* bias[3]   = { (const float*)d_in[10], (const float*)d_in[16], (const float*)d_in[22] };
    const float* fc1_W     = (const float*)d_in[23];
    const float* fc1_b     = (const float*)d_in[24];
    const float* fc2_W     = (const float*)d_in[25];
    const float* fc2_b     = (const float*)d_in[26];
    float* out             = (float*)d_out;
    const int HEADS[3]     = {4, 4, 1};

    // ---- workspace bump allocator (256B aligned) ----
    size_t off = 0;
    auto alloc = [&](size_t nfloats) -> float* {
        float* p = (float*)((char*)d_ws + off);
        off += ((nfloats * sizeof(float) + 255) / 256) * 256;
        return p;
    };
    float*    deg       = alloc((size_t)NN);
    float*    loop_attr = alloc((size_t)NN * 2);
    float*    hA        = alloc((size_t)NN * DD);
    float*    hB        = alloc((size_t)NN * DD);
    float*    xw        = alloc((size_t)NN * 256);
    float*    a_src     = alloc((size_t)NN * 4);
    float*    a_dst     = alloc((size_t)NN * 4);
    float*    logits    = alloc((size_t)ETOT * 4);
    unsigned* menc      = (unsigned*)alloc((size_t)NN * 4);
    float*    den       = alloc((size_t)NN * 4);
    float*    outbuf    = alloc((size_t)NN * 256);
    float*    coef      = alloc(16);
    float*    h2        = alloc((size_t)NN * 32);
    (void)ws_size; (void)n_in; (void)in_sizes; (void)out_size;

    const int BS = 256;

    // ---- self-loop attrs (mean of incoming edge_attr) ----
    fill_f32<<<gridFor(NN, BS), BS, 0, stream>>>(deg, 0.0f, (size_t)NN);
    fill_f32<<<gridFor((long long)NN * 2, BS), BS, 0, stream>>>(loop_attr, 0.0f, (size_t)NN * 2);
    deg_sum_k<<<gridFor(EE, BS), BS, 0, stream>>>(ei, ea, deg, loop_attr);
    loop_fin_k<<<gridFor(NN, BS), BS, 0, stream>>>(loop_attr, deg);

    // ---- encoder ----
    encoder_k<<<gridFor((long long)NN * DD, BS), BS, 0, stream>>>(x, enc_W, enc_b, hA);

    float* hcur = hA;
    float* hnext = hB;

    // ---- 3 GAT layers ----
    for (int L = 0; L < 3; ++L) {
        int H = HEADS[L];
        int M = H * DD;
        int colTiles = M / 16;
        int totalTiles = (NN / 16) * colTiles;  // N divisible by 16

        // xw = h @ lin_W   (WMMA fp32)
        gemm_wmma_f32<<<gridFor((long long)totalTiles * 32, BS), BS, 0, stream>>>(
            hcur, lin_W[L], xw, nullptr, M, colTiles, totalTiles, 0);

        // attention scalars
        att_node_k<<<gridFor((long long)NN * H, BS), BS, 0, stream>>>(
            xw, att_src[L], att_dst[L], a_src, a_dst, H);
        edge_coef_k<<<1, 64, 0, stream>>>(edge_W[L], att_edge[L], coef, H);

        // reset accumulators
        fill_u32<<<gridFor((long long)NN * H, BS), BS, 0, stream>>>(menc, 0u, (size_t)NN * H);
        fill_f32<<<gridFor((long long)NN * H, BS), BS, 0, stream>>>(den, 0.0f, (size_t)NN * H);
        fill_f32<<<gridFor((long long)NN * M, BS), BS, 0, stream>>>(outbuf, 0.0f, (size_t)NN * M);

        // segment softmax + aggregate
        edge_logits_k<<<gridFor(ETOT, BS), BS, 0, stream>>>(
            ei, ea, loop_attr, a_src, a_dst, coef, logits, menc, H);
        edge_exp_k<<<gridFor(ETOT, BS), BS, 0, stream>>>(ei, logits, menc, den, H);
        edge_aggr_k<<<gridFor((long long)ETOT * H, BS), BS, 0, stream>>>(
            ei, logits, den, xw, outbuf, H);

        // mean over heads + bias + relu + residual
        layer_fin_k<<<gridFor((long long)NN * DD, BS), BS, 0, stream>>>(
            outbuf, bias[L], hcur, hnext, H);

        float* tmp = hcur; hcur = hnext; hnext = tmp;
    }

    // ---- fc1 (WMMA, fused bias+relu) ----
    {
        int M = 32, colTiles = 2, totalTiles = (NN / 16) * colTiles;
        gemm_wmma_f32<<<gridFor((long long)totalTiles * 32, BS), BS, 0, stream>>>(
            hcur, fc1_W, h2, fc1_b, M, colTiles, totalTiles, 1);
    }
    // ---- fc2 + tanh ----
    fc2_k<<<gridFor(NN, BS), BS, 0, stream>>>(h2, fc2_W, fc2_b, out);
}